// JKNet_47278999994739
// MI455X (gfx1250) — compile-verified
//
#include <hip/hip_runtime.h>

typedef float v2f __attribute__((ext_vector_type(2)));
typedef float v8f __attribute__((ext_vector_type(8)));

#define HID 64

// Native GLOBAL_ATOMIC_ADD_F32 (no CAS expansion). unsafeAtomicAdd lowers to
// the hardware f32 fadd atomic on AMDGCN targets.
__device__ __forceinline__ void atomic_add_f32(float* p, float v) {
  unsafeAtomicAdd(p, v);
}

__global__ void zero_f32(float* __restrict__ p, int n) {
  int i = blockIdx.x * blockDim.x + threadIdx.x;
  if (i < n) p[i] = 0.0f;
}

__global__ void init_out_kernel(float* __restrict__ out, const float* __restrict__ bout, int total) {
  int i = blockIdx.x * blockDim.x + threadIdx.x;
  if (i < total) out[i] = bout[i & (HID - 1)];
}

__global__ void degree_kernel(const int* __restrict__ src, const int* __restrict__ dst,
                              float* __restrict__ out_deg, float* __restrict__ in_deg, int E) {
  int e = blockIdx.x * blockDim.x + threadIdx.x;
  if (e < E) {
    atomic_add_f32(&out_deg[src[e]], 1.0f);
    atomic_add_f32(&in_deg[dst[e]], 1.0f);
  }
}

__global__ void rsqrt_inplace(float* __restrict__ deg, int n) {
  int i = blockIdx.x * blockDim.x + threadIdx.x;
  if (i < n) deg[i] = __frsqrt_rn(fmaxf(deg[i], 1.0f));
}

// out[dst[e], :] += feat[src[e], :]   (64 features, float4 per thread)
__global__ void scatter_add64(const float* __restrict__ feat,
                              const int* __restrict__ src,
                              const int* __restrict__ dst,
                              float* __restrict__ out, int E) {
  int idx = blockIdx.x * blockDim.x + threadIdx.x; // E*16 work units
  if (idx >= E * 16) return;
  int e = idx >> 4;
  int q = (idx & 15) << 2;
  const float4 v = *(const float4*)(feat + (long)src[e] * HID + q);
  float* o = out + (long)dst[e] * HID + q;
  atomic_add_f32(o + 0, v.x);
  atomic_add_f32(o + 1, v.y);
  atomic_add_f32(o + 2, v.z);
  atomic_add_f32(o + 3, v.w);
}

// h = relu(agg * norm_dst[row] + bias[col])
__global__ void relu_bias_norm(const float* __restrict__ agg,
                               const float* __restrict__ ndst,
                               const float* __restrict__ bias,
                               float* __restrict__ h, int total) {
  int idx = blockIdx.x * blockDim.x + threadIdx.x;
  if (idx < total) {
    int row = idx >> 6;
    int col = idx & 63;
    float v = agg[idx] * ndst[row] + bias[col];
    h[idx] = fmaxf(v, 0.0f);
  }
}

// out[nrows x 64] (=|+=) (A[nrows x K] * rowscale) @ W[K x 64]
// One wave computes a 16x64 tile via V_WMMA_F32_16X16X4_F32; W staged in LDS.
// Requires nrows % 16 == 0 (true here: 100000 = 16 * 6250).
__global__ void gemm_wmma_f32(const float* __restrict__ A,
                              const float* __restrict__ rowscale,
                              const float* __restrict__ W,
                              float* __restrict__ out,
                              int nrows, int K, int accumulate) {
  extern __shared__ float sW[]; // K x 64
  for (int i = threadIdx.x; i < K * HID; i += blockDim.x) sW[i] = W[i];
  __syncthreads();

  const int wave = threadIdx.x >> 5;
  const int lane = threadIdx.x & 31;
  const int r0 = (blockIdx.x * 8 + wave) * 16;
  if (r0 >= nrows) return; // wave-uniform: EXEC all-ones for WMMA below

  const int half = lane >> 4;  // 0 or 1
  const int l16 = lane & 15;
  const int koff = half * 2;   // lanes 0-15: K={k,k+1}; lanes 16-31: K={k+2,k+3}

  const int arow = r0 + l16;
  const float scale = rowscale ? rowscale[arow] : 1.0f;
  const float* aptr = A + (long)arow * K;

  v8f acc0 = {}, acc1 = {}, acc2 = {}, acc3 = {};

  for (int k = 0; k < K; k += 4) {
    v2f a;
    a.x = aptr[k + koff] * scale;
    a.y = aptr[k + koff + 1] * scale;
    const float* w0 = sW + (k + koff) * HID + l16;
    const float* w1 = sW + (k + koff + 1) * HID + l16;
    v2f b0; b0.x = w0[0];  b0.y = w1[0];
    v2f b1; b1.x = w0[16]; b1.y = w1[16];
    v2f b2; b2.x = w0[32]; b2.y = w1[32];
    v2f b3; b3.x = w0[48]; b3.y = w1[48];
    acc0 = __builtin_amdgcn_wmma_f32_16x16x4_f32(false, a, false, b0, (short)0, acc0, false, false);
    acc1 = __builtin_amdgcn_wmma_f32_16x16x4_f32(false, a, false, b1, (short)0, acc1, false, false);
    acc2 = __builtin_amdgcn_wmma_f32_16x16x4_f32(false, a, false, b2, (short)0, acc2, false, false);
    acc3 = __builtin_amdgcn_wmma_f32_16x16x4_f32(false, a, false, b3, (short)0, acc3, false, false);
  }

  // C/D layout: VGPR i, lanes 0-15 -> row i, lanes 16-31 -> row i+8; col = lane&15 (+16*tile)
  float* obase = out + (long)r0 * HID;
#pragma unroll
  for (int i = 0; i < 8; ++i) {
    float* op = obase + (long)(i + half * 8) * HID + l16;
    if (accumulate) {
      op[0]  += acc0[i];
      op[16] += acc1[i];
      op[32] += acc2[i];
      op[48] += acc3[i];
    } else {
      op[0]  = acc0[i];
      op[16] = acc1[i];
      op[32] = acc2[i];
      op[48] = acc3[i];
    }
  }
}

static inline int cdiv(int a, int b) { return (a + b - 1) / b; }

extern "C" void kernel_launch(void* const* d_in, const int* in_sizes, int n_in,
                              void* d_out, int out_size, void* d_ws, size_t ws_size,
                              hipStream_t stream) {
  const float* feats = (const float*)d_in[0];
  const int* src = (const int*)d_in[1];
  const int* dst = (const int*)d_in[2];
  const float* W[5] = {(const float*)d_in[3], (const float*)d_in[5], (const float*)d_in[7],
                       (const float*)d_in[9], (const float*)d_in[11]};
  const float* b[5] = {(const float*)d_in[4], (const float*)d_in[6], (const float*)d_in[8],
                       (const float*)d_in[10], (const float*)d_in[12]};
  const float* Wout = (const float*)d_in[13];
  const float* bout = (const float*)d_in[14];

  const int E = in_sizes[1];
  const int N = out_size / HID;           // 100000
  const int K0 = in_sizes[0] / N;         // 256

  float* ws = (float*)d_ws;
  float* out_norm = ws; ws += N;
  float* in_norm  = ws; ws += N;
  float* h   = ws; ws += (size_t)N * HID;
  float* hs  = ws; ws += (size_t)N * HID;
  float* agg = ws; ws += (size_t)N * HID;

  float* out = (float*)d_out;

  // degrees -> rsqrt norms
  zero_f32<<<cdiv(N, 256), 256, 0, stream>>>(out_norm, N);
  zero_f32<<<cdiv(N, 256), 256, 0, stream>>>(in_norm, N);
  degree_kernel<<<cdiv(E, 256), 256, 0, stream>>>(src, dst, out_norm, in_norm, E);
  rsqrt_inplace<<<cdiv(N, 256), 256, 0, stream>>>(out_norm, N);
  rsqrt_inplace<<<cdiv(N, 256), 256, 0, stream>>>(in_norm, N);

  // out = bout (broadcast); final GEMMs accumulate onto it
  init_out_kernel<<<cdiv(N * HID, 256), 256, 0, stream>>>(out, bout, N * HID);

  const int gemmBlocks = cdiv(cdiv(N, 16), 8);
  const int nh = N * HID;

  for (int i = 0; i < 5; ++i) {
    const float* inp = (i == 0) ? feats : h;
    const int K = (i == 0) ? K0 : HID;

    // hs = (inp * norm_src) @ W_i
    gemm_wmma_f32<<<gemmBlocks, 256, (size_t)K * HID * sizeof(float), stream>>>(
        inp, out_norm, W[i], hs, N, K, 0);

    // agg = segment_sum(hs[src], dst)
    zero_f32<<<cdiv(nh, 256), 256, 0, stream>>>(agg, nh);
    scatter_add64<<<cdiv(E * 16, 256), 256, 0, stream>>>(hs, src, dst, agg, E);

    // h = relu(agg * norm_dst + b_i)
    relu_bias_norm<<<cdiv(nh, 256), 256, 0, stream>>>(agg, in_norm, b[i], h, nh);

    // hfin_i = segment_sum(h[src], dst)   (reuse agg)
    zero_f32<<<cdiv(nh, 256), 256, 0, stream>>>(agg, nh);
    scatter_add64<<<cdiv(E * 16, 256), 256, 0, stream>>>(h, src, dst, agg, E);

    // out += hfin_i @ Wout[64*i : 64*i+64, :]
    gemm_wmma_f32<<<gemmBlocks, 256, (size_t)HID * HID * sizeof(float), stream>>>(
        agg, nullptr, Wout + (size_t)i * HID * HID, out, N, HID, 1);
  }
}